// Aggregator_72567767433687
// MI455X (gfx1250) — compile-verified
//
#include <hip/hip_runtime.h>
#include <hip/hip_bf16.h>

// ---------------------------------------------------------------------------
// KGAT-style aggregator for MI455X (gfx1250, wave32).
//   pass A: batched dot scores via V_WMMA_F32_16X16X4_F32 (exact f32 FMA);
//           rows gathered straight into LDS with GLOBAL_LOAD_ASYNC_TO_LDS_B128
//           (ASYNCcnt path, no VGPR round-trip); atomic segment max
//   pass B: exp(x - max), atomic segment sum
//   pass C: normalize + scatter-add rows with global_atomic_add_f32
// Workspace layout (floats): w1raw[E] | attraw[I] | emax[NE] | esum[NE] |
//                            umax[NU] | usum[NU]   (~10.4 MB)
// ---------------------------------------------------------------------------

typedef __attribute__((ext_vector_type(2))) float v2f;
typedef __attribute__((ext_vector_type(8))) float v8f;

#define DIMV 64
#define LSTRIDE 68      // floats; 272B rows: 16B aligned (async B128) and
                        // bank shift 4/row -> 16 fragment rows on distinct banks
#define WAVES 4         // waves per block in score kernels (128 threads)

__device__ __forceinline__ void atomicMaxF(float* addr, float v) {
    // order-preserving float max via integer atomics (init must be -inf)
    if (v >= 0.0f) atomicMax((int*)addr, __float_as_int(v));
    else           atomicMin((unsigned int*)addr, __float_as_uint(v));
}

// Async gather: 16B global -> LDS, tracked by ASYNCcnt (per-lane addresses).
__device__ __forceinline__ void async_copy_b128(unsigned lds_off, const void* gptr) {
    asm volatile("global_load_async_to_lds_b128 %0, %1, off"
                 :: "v"(lds_off), "v"(gptr) : "memory");
}
__device__ __forceinline__ void wait_async0() {
    asm volatile("s_wait_asynccnt 0x0" ::: "memory");
}
// generic->LDS offset: low 32 bits of the flat address are the group-segment
// offset on gfx1250 (aperture lives in the high bits)
__device__ __forceinline__ unsigned lds_off32(const void* p) {
    return (unsigned)(unsigned long long)p;
}

// ---------------------------------------------------------------------------
// Init: zero output accumulators, set segment max = -inf, segment sum = 0.
// ---------------------------------------------------------------------------
__global__ void init_kernel(float* __restrict__ out, long nOut,
                            float* __restrict__ emax, float* __restrict__ esum, int NE,
                            float* __restrict__ umax, float* __restrict__ usum, int NU) {
    const long i0 = (long)blockIdx.x * blockDim.x + threadIdx.x;
    const long stride = (long)gridDim.x * blockDim.x;
    const float NEG_INF = __int_as_float((int)0xff800000);
    for (long j = i0; j < nOut; j += stride) out[j] = 0.0f;
    for (long j = i0; j < NE;   j += stride) { emax[j] = NEG_INF; esum[j] = 0.0f; }
    for (long j = i0; j < NU;   j += stride) { umax[j] = NEG_INF; usum[j] = 0.0f; }
}

// ---------------------------------------------------------------------------
// WMMA core: staged A,B (16 rows x 64 cols, stride LSTRIDE) in LDS ->
// diag(A x B^T) with 16 chained f32 16x16x4 WMMAs (exact f32 FMA, RNE).
// Fragment layout (ISA 7.12.2, wave32): both A 16x4 and B 4x16 reduce to
// reading float2 at [ (lane&15)*LSTRIDE + kk*4 + (lane>>4)*2 ].
// ---------------------------------------------------------------------------
__device__ __forceinline__ float wmma_diag_dot(const float* lA, const float* lB, int lid) {
    const int frow = lid & 15;
    const int koff = (lid >> 4) * 2;
    v8f c = {};
#pragma unroll
    for (int kk = 0; kk < 16; ++kk) {
        const float2 af = *(const float2*)&lA[frow * LSTRIDE + kk * 4 + koff];
        const float2 bf = *(const float2*)&lB[frow * LSTRIDE + kk * 4 + koff];
        v2f a; a.x = af.x; a.y = af.y;
        v2f b; b.x = bf.x; b.y = bf.y;
        c = __builtin_amdgcn_wmma_f32_16x16x4_f32(false, a, false, b, (short)0, c,
                                                  false, false);
    }
    // diag element (i,i): i<8 -> lane i, VGPR i ; i>=8 -> lane i+16, VGPR i-8
    const int i = lid & 15;
    const int srcLane = (i < 8) ? i : (i + 16);
    const int k = i & 7;
    float diag = 0.0f;
#pragma unroll
    for (int kk = 0; kk < 8; ++kk) {
        float t = __shfl(c[kk], srcLane, 32);
        if (kk == k) diag = t;
    }
    return diag;
}

// ---------------------------------------------------------------------------
// Pass A (entity): 16 edges per wave; async-gather neigh + relation rows into
// LDS, WMMA Gram tile, diagonal/8, atomic segment max over head.
// ---------------------------------------------------------------------------
__global__ __launch_bounds__(128) void edge_score_wmma(
        const float* __restrict__ entity_emb, const float* __restrict__ rel_emb,
        const int* __restrict__ edge_index, const int* __restrict__ edge_type,
        float* __restrict__ w1raw, float* __restrict__ emax, int E) {
    __shared__ float lA[WAVES][16 * LSTRIDE];
    __shared__ float lB[WAVES][16 * LSTRIDE];
    __shared__ int   lTail[WAVES][16];
    __shared__ int   lRel [WAVES][16];

    const int w = threadIdx.x >> 5;
    const int lid = threadIdx.x & 31;
    const long tileBase = ((long)blockIdx.x * WAVES + w) * 16;

    if (lid < 16) {
        long e = tileBase + lid; if (e >= E) e = E - 1;   // clamp: EXEC stays full later
        lTail[w][lid] = edge_index[(long)E + e];           // row 1 = tail
        int r = edge_type[e] - 1;
        if (r < 0) r = 0; if (r > 32) r = 32;
        lRel[w][lid] = r;
    }
    __syncthreads();

    const unsigned ldsA = lds_off32(&lA[w][0]);
    const unsigned ldsB = lds_off32(&lB[w][0]);
#pragma unroll
    for (int it = 0; it < 8; ++it) {                       // 256 x 16B chunks / matrix
        const int c = it * 32 + lid;
        const int row = c >> 4, c4 = c & 15;
        const unsigned loff = (unsigned)(row * LSTRIDE + c4 * 4) * 4u;
        async_copy_b128(ldsA + loff, entity_emb + (long)lTail[w][row] * DIMV + c4 * 4);
        async_copy_b128(ldsB + loff, rel_emb   + (long)lRel [w][row] * DIMV + c4 * 4);
    }
    wait_async0();
    __syncthreads();

    const float diag = wmma_diag_dot(lA[w], lB[w], lid);

    if (lid < 16) {
        const long e = tileBase + lid;
        if (e < E) {
            const float s = diag * 0.125f;                 // /8
            w1raw[e] = s;
            atomicMaxF(&emax[edge_index[e]], s);           // row 0 = head
        }
    }
}

// ---------------------------------------------------------------------------
// Pass A (user): A = entity[item] (async gather), B = inter[t] ⊙ user[u]
// (register path, needs the product).
// ---------------------------------------------------------------------------
__global__ __launch_bounds__(128) void user_score_wmma(
        const float* __restrict__ entity_emb, const float* __restrict__ user_emb,
        const float* __restrict__ inter_emb,
        const int* __restrict__ user_index, const int* __restrict__ item_index,
        const int* __restrict__ itype,
        float* __restrict__ attraw, float* __restrict__ umax, int I) {
    __shared__ float lA[WAVES][16 * LSTRIDE];
    __shared__ float lB[WAVES][16 * LSTRIDE];
    __shared__ int   lItem[WAVES][16];
    __shared__ int   lUsr [WAVES][16];
    __shared__ int   lTyp [WAVES][16];

    const int w = threadIdx.x >> 5;
    const int lid = threadIdx.x & 31;
    const long tileBase = ((long)blockIdx.x * WAVES + w) * 16;

    if (lid < 16) {
        long i = tileBase + lid; if (i >= I) i = I - 1;
        lItem[w][lid] = item_index[i];
        lUsr [w][lid] = user_index[i];
        int t = itype[i]; if (t < 0) t = 0; if (t > 7) t = 7;
        lTyp [w][lid] = t;
    }
    __syncthreads();

    const unsigned ldsA = lds_off32(&lA[w][0]);
#pragma unroll
    for (int it = 0; it < 8; ++it) {
        const int c = it * 32 + lid;
        const int row = c >> 4, c4 = c & 15;
        // A rows: straight async gather into LDS
        async_copy_b128(ldsA + (unsigned)(row * LSTRIDE + c4 * 4) * 4u,
                        entity_emb + (long)lItem[w][row] * DIMV + c4 * 4);
        // B rows: elementwise product, via registers (16B-aligned store)
        const float4 vu = ((const float4*)(user_emb  + (long)lUsr[w][row] * DIMV))[c4];
        const float4 vt = ((const float4*)(inter_emb + (long)lTyp[w][row] * DIMV))[c4];
        *(float4*)&lB[w][row * LSTRIDE + c4 * 4] =
            make_float4(vu.x * vt.x, vu.y * vt.y, vu.z * vt.z, vu.w * vt.w);
    }
    wait_async0();
    __syncthreads();

    const float diag = wmma_diag_dot(lA[w], lB[w], lid);

    if (lid < 16) {
        const long i = tileBase + lid;
        if (i < I) {
            attraw[i] = diag;
            atomicMaxF(&umax[lUsr[w][lid]], diag);
        }
    }
}

// ---------------------------------------------------------------------------
// Pass B (both): exp(raw - seg_max), in-place, atomic segment sum.
// ---------------------------------------------------------------------------
__global__ void seg_exp_sum(const int* __restrict__ seg, const float* __restrict__ mx,
                            float* __restrict__ raw, float* __restrict__ ssum, int n) {
    const long i = (long)blockIdx.x * blockDim.x + threadIdx.x;
    if (i >= n) return;
    const int s = seg[i];
    const float ex = __expf(raw[i] - mx[s]);
    raw[i] = ex;
    atomicAdd(&ssum[s], ex);
}

// ---------------------------------------------------------------------------
// Pass C (entity): w = exp/sum(head); out_w1[e] = w; entity_agg[head] += w*neigh.
// 64 threads per edge (one per feature dim), 4 edges per 256-thread block.
// ---------------------------------------------------------------------------
__global__ void edge_aggregate(const float* __restrict__ entity_emb,
                               const int* __restrict__ edge_index,
                               const float* __restrict__ expv, const float* __restrict__ esum,
                               float* __restrict__ out_entity, float* __restrict__ out_w1,
                               int E) {
    const long e = (long)blockIdx.x * 4 + (threadIdx.x >> 6);
    const int d = threadIdx.x & 63;
    if (e >= E) return;
    const int head = edge_index[e];
    const int tail = edge_index[(long)E + e];
    const float wgt = expv[e] / esum[head];
    if (d == 0) out_w1[e] = wgt;
    const float v = entity_emb[(long)tail * DIMV + d];
    atomicAdd(&out_entity[(long)head * DIMV + d], v * wgt);
}

// ---------------------------------------------------------------------------
// Pass C (user): w = exp/sum(user); out_att[i] = w; user_agg[u] += w*item_row.
// ---------------------------------------------------------------------------
__global__ void user_aggregate(const float* __restrict__ entity_emb,
                               const int* __restrict__ user_index,
                               const int* __restrict__ item_index,
                               const float* __restrict__ expv, const float* __restrict__ usum,
                               float* __restrict__ out_user, float* __restrict__ out_att,
                               int I) {
    const long i = (long)blockIdx.x * 4 + (threadIdx.x >> 6);
    const int d = threadIdx.x & 63;
    if (i >= I) return;
    const int u = user_index[i];
    const int it = item_index[i];
    const float wgt = expv[i] / usum[u];
    if (d == 0) out_att[i] = wgt;
    const float v = entity_emb[(long)it * DIMV + d];
    atomicAdd(&out_user[(long)u * DIMV + d], v * wgt);
}

// ---------------------------------------------------------------------------
extern "C" void kernel_launch(void* const* d_in, const int* in_sizes, int n_in,
                              void* d_out, int out_size, void* d_ws, size_t ws_size,
                              hipStream_t stream) {
    const float* entity_emb = (const float*)d_in[0];
    const float* user_emb   = (const float*)d_in[1];
    const float* inter_emb  = (const float*)d_in[2];
    const float* rel_emb    = (const float*)d_in[3];
    const int*   edge_index = (const int*)d_in[4];
    const int*   edge_type  = (const int*)d_in[5];
    const int*   user_index = (const int*)d_in[6];
    const int*   item_index = (const int*)d_in[7];
    const int*   itype      = (const int*)d_in[8];

    const int NE = in_sizes[0] / DIMV;
    const int NU = in_sizes[1] / DIMV;
    const int E  = in_sizes[5];
    const int I  = in_sizes[6];

    float* out        = (float*)d_out;
    float* out_entity = out;                                  // NE*64
    float* out_user   = out + (long)NE * DIMV;                // NU*64
    float* out_att    = out_user + (long)NU * DIMV;           // I
    float* out_w1     = out_att + I;                          // E

    float* wsf    = (float*)d_ws;
    float* w1raw  = wsf;                                      // E
    float* attraw = w1raw + E;                                // I
    float* emax   = attraw + I;                               // NE
    float* esum   = emax + NE;                                // NE
    float* umax   = esum + NE;                                // NU
    float* usum   = umax + NU;                                // NU

    // 0) init accumulators / segment state (harness does not re-zero buffers)
    init_kernel<<<4096, 256, 0, stream>>>(out, (long)(NE + NU) * DIMV,
                                          emax, esum, NE, umax, usum, NU);

    // ---- entity pipeline ----
    edge_score_wmma<<<(E + WAVES * 16 - 1) / (WAVES * 16), WAVES * 32, 0, stream>>>(
        entity_emb, rel_emb, edge_index, edge_type, w1raw, emax, E);
    seg_exp_sum<<<(E + 255) / 256, 256, 0, stream>>>(edge_index, emax, w1raw, esum, E);
    edge_aggregate<<<(E + 3) / 4, 256, 0, stream>>>(entity_emb, edge_index, w1raw, esum,
                                                    out_entity, out_w1, E);

    // ---- user pipeline ----
    user_score_wmma<<<(I + WAVES * 16 - 1) / (WAVES * 16), WAVES * 32, 0, stream>>>(
        entity_emb, user_emb, inter_emb, user_index, item_index, itype, attraw, umax, I);
    seg_exp_sum<<<(I + 255) / 256, 256, 0, stream>>>(user_index, umax, attraw, usum, I);
    user_aggregate<<<(I + 3) / 4, 256, 0, stream>>>(entity_emb, user_index, item_index,
                                                    attraw, usum, out_user, out_att, I);
}